// DifferentiableMPCController_14413910246013
// MI455X (gfx1250) — compile-verified
//
#include <hip/hip_runtime.h>

// Batched LQR (Riccati backward + closed-loop rollout) for gfx1250.
// One wave32 per batch item; fp32 WMMA (V_WMMA_F32_16X16X4_F32) for all
// tile matmuls; LDS-staged operands; lane-parallel 16x16 SPD solve.
// Next step's cost matrix C_t is double-buffered into LDS with
// GLOBAL_LOAD_ASYNC_TO_LDS_B128 (ASYNCcnt) to hide HBM latency on the
// serial per-wave recursion.

#define NXd 32
#define NUd 16
#define NNd 48   // NX + NU
#define HHd 50
#define BBd 1024

#if defined(__has_builtin)
#if __has_builtin(__builtin_amdgcn_global_load_async_to_lds_b128)
#define LQR_ASYNC 1
#endif
#endif
#ifndef LQR_ASYNC
#define LQR_ASYNC 0
#endif

typedef __attribute__((ext_vector_type(2))) float v2f;
typedef __attribute__((ext_vector_type(8))) float v8f;
typedef __attribute__((ext_vector_type(4))) int   v4i;

#if LQR_ASYNC
typedef __attribute__((address_space(3))) v4i lv4i_t;

__device__ __forceinline__ void async_cp_b128(const char* gsrc, char* ldst) {
  // builtin signature (from clang diagnostic): param0 = generic v4i* (payload
  // address in global memory), param1 = LDS destination, imm offset, imm cpol.
  __builtin_amdgcn_global_load_async_to_lds_b128(
      (v4i*)gsrc, (lv4i_t*)ldst, 0, 0);
}
#endif

__device__ __forceinline__ v8f wmma_f32(v2f a, v2f b, v8f c) {
  // 8 args: (neg_a, A, neg_b, B, c_mod, C, reuse_a, reuse_b)
  return __builtin_amdgcn_wmma_f32_16x16x4_f32(false, a, false, b, (short)0, c,
                                               false, false);
}

__global__ __launch_bounds__(32)
void lqr_kernel(const float* __restrict__ x0g,
                const float* __restrict__ Cg,
                const float* __restrict__ cg,
                const float* __restrict__ Cfg,
                const float* __restrict__ cfg,
                const float* __restrict__ Ag,
                const float* __restrict__ Bmg,
                float* __restrict__ outg,
                float* __restrict__ wsK,   // [B][H][NU][NX] gains
                float* __restrict__ wsk)   // [B][H][NU] offsets
{
  __shared__ float sFT [NNd * NXd];  // F^T (48x32)
  __shared__ float sF  [NXd * NNd];  // F = [A | Bm] (32x48)
  __shared__ float sV  [NXd * NXd];
  __shared__ float sVn [NXd * NXd];
  __shared__ float sFTV[NNd * NXd];
  __shared__ float sQ  [NNd * NNd];
  __shared__ float sW  [16 * 52];    // augmented [Quu | Qxu^T | qu], 49 cols
  __shared__ float sSol[16 * 36];    // 33 cols
  __shared__ float sK  [NUd * NXd];
  __shared__ float sq  [NNd];
  __shared__ float sv  [NXd];
  __shared__ float skv [NUd];
  __shared__ float sxu [NNd];
#if LQR_ASYNC
  __shared__ float sCt [2][NNd * NNd];  // double-buffered cost matrix tile
#endif

  const int b    = blockIdx.x;
  const int lane = threadIdx.x;
  const int mlo  = lane & 15;        // row (A) or col (B/C) index within tile
  const int khi  = (lane >> 4) * 2;  // K sub-offset for A/B frags
  const int hi8  = (lane >> 4) * 8;  // M sub-offset for C/D frags

  // ---- stage constants: F = [A | Bm] and F^T ----
  for (int idx = lane; idx < NXd * NNd; idx += 32) {
    int x = idx / NNd, i = idx % NNd;
    float val = (i < NXd) ? Ag[x * NXd + i] : Bmg[x * NUd + (i - NXd)];
    sF[x * NNd + i]  = val;
    sFT[i * NXd + x] = val;
  }
  // ---- V0 = C_final[:nx,:nx], v0 = c_final[:nx] ----
  for (int idx = lane; idx < NXd * NXd; idx += 32) {
    int i = idx >> 5, j = idx & 31;
    sV[idx] = Cfg[(size_t)b * NNd * NNd + i * NNd + j];
  }
  sv[lane] = cfg[(size_t)b * NNd + lane];

#if LQR_ASYNC
  // ---- prime buffer 0 with C_t for s=0 (time H-1): 18 x 512B wave-wide ----
  {
    const char* src = (const char*)(Cg + ((size_t)b * HHd + (HHd - 1)) * NNd * NNd);
    char* dst = (char*)&sCt[0][0];
    const int off = lane * 16;
#pragma unroll
    for (int i = 0; i < 18; ++i)
      async_cp_b128(src + i * 512 + off, dst + i * 512 + off);
  }
#endif
  __syncthreads();

  // ================= backward Riccati recursion =================
  for (int s = 0; s < HHd; ++s) {
    const int tcur = HHd - 1 - s;  // forward-time index of this step's gains
    const float* ct = cg + ((size_t)b * HHd + tcur) * NNd;

#if LQR_ASYNC
    const int cur = s & 1;
    if (s + 1 < HHd) {  // kick off async copy of next step's C_t
      const char* src = (const char*)(Cg + ((size_t)b * HHd + (tcur - 1)) * NNd * NNd);
      char* dst = (char*)&sCt[cur ^ 1][0];
      const int off = lane * 16;
#pragma unroll
      for (int i = 0; i < 18; ++i)
        async_cp_b128(src + i * 512 + off, dst + i * 512 + off);
    }
#else
    const float* Ct = Cg + ((size_t)b * HHd + tcur) * NNd * NNd;
    if (s + 1 < HHd) {  // prefetch next step's cost matrix (~9KB)
      const char* nxt = (const char*)(Cg + ((size_t)b * HHd + (tcur - 1)) * NNd * NNd);
      __builtin_prefetch(nxt + lane * 288, 0, 0);
    }
#endif

    // ---- FTV = F^T * V : (48x32) = (48x32)(32x32), 3x2 tiles, K=32 ----
    for (int mt = 0; mt < 3; ++mt)
      for (int nt = 0; nt < 2; ++nt) {
        v8f acc = {0.f, 0.f, 0.f, 0.f, 0.f, 0.f, 0.f, 0.f};
#pragma unroll
        for (int kc = 0; kc < 8; ++kc) {
          v2f a, bb;
          a.x  = sFT[(mt * 16 + mlo) * NXd + kc * 4 + khi];
          a.y  = sFT[(mt * 16 + mlo) * NXd + kc * 4 + khi + 1];
          bb.x = sV[(kc * 4 + khi) * NXd + nt * 16 + mlo];
          bb.y = sV[(kc * 4 + khi + 1) * NXd + nt * 16 + mlo];
          acc = wmma_f32(a, bb, acc);
        }
#pragma unroll
        for (int i = 0; i < 8; ++i)
          sFTV[(mt * 16 + i + hi8) * NXd + nt * 16 + mlo] = acc[i];
      }
    __syncthreads();

#if LQR_ASYNC
    // async loads complete in order: with 18 newer ops in flight, cnt<=18
    // guarantees the older batch (this step's buffer) has landed in LDS.
    if (s + 1 < HHd) {
      asm volatile("s_wait_asynccnt 18" ::: "memory");
    } else {
      asm volatile("s_wait_asynccnt 0" ::: "memory");
    }
#endif

    // ---- Q = Ct + FTV * F : (48x48), 3x3 tiles, K=32 ----
    for (int mt = 0; mt < 3; ++mt)
      for (int nt = 0; nt < 3; ++nt) {
        v8f acc;
#pragma unroll
        for (int i = 0; i < 8; ++i)
#if LQR_ASYNC
          acc[i] = sCt[cur][(mt * 16 + i + hi8) * NNd + nt * 16 + mlo];
#else
          acc[i] = Ct[(mt * 16 + i + hi8) * NNd + nt * 16 + mlo];
#endif
#pragma unroll
        for (int kc = 0; kc < 8; ++kc) {
          v2f a, bb;
          a.x  = sFTV[(mt * 16 + mlo) * NXd + kc * 4 + khi];
          a.y  = sFTV[(mt * 16 + mlo) * NXd + kc * 4 + khi + 1];
          bb.x = sF[(kc * 4 + khi) * NNd + nt * 16 + mlo];
          bb.y = sF[(kc * 4 + khi + 1) * NNd + nt * 16 + mlo];
          acc = wmma_f32(a, bb, acc);
        }
#pragma unroll
        for (int i = 0; i < 8; ++i)
          sQ[(mt * 16 + i + hi8) * NNd + nt * 16 + mlo] = acc[i];
      }

    // ---- q = ct + F^T v (lane-parallel matvec) ----
    for (int i = lane; i < NNd; i += 32) {
      float sacc = ct[i];
#pragma unroll
      for (int x = 0; x < NXd; ++x) sacc += sFT[i * NXd + x] * sv[x];
      sq[i] = sacc;
    }
    __syncthreads();

    // ---- build augmented system [Quu+eps*I | Qxu^T | qu] (16 x 49) ----
    for (int idx = lane; idx < 16 * 49; idx += 32) {
      int r = idx / 49, cc = idx % 49;
      float val;
      if (cc < 16)
        val = sQ[(NXd + r) * NNd + NXd + cc] + (r == cc ? 1e-6f : 0.0f);
      else if (cc < 48)
        val = sQ[(cc - 16) * NNd + NXd + r];  // Qxu^T[r][j] = Q[j][32+r]
      else
        val = sq[NXd + r];
      sW[r * 52 + cc] = val;
    }
    __syncthreads();

    // ---- Gaussian elimination (SPD + eps, no pivoting) ----
    for (int k = 0; k < 16; ++k) {
      float invp = 1.0f / sW[k * 52 + k];
      for (int r = k + 1; r < 16; ++r) {
        float f = sW[r * 52 + k] * invp;
        for (int cc = k + 1 + lane; cc < 49; cc += 32)
          sW[r * 52 + cc] -= f * sW[k * 52 + cc];
      }
      __syncthreads();
    }

    // ---- back substitution: 33 RHS columns across lanes ----
    for (int cc2 = lane; cc2 < 33; cc2 += 32) {
      float sol[16];
#pragma unroll
      for (int r = 15; r >= 0; --r) {
        float sacc = sW[r * 52 + 16 + cc2];
#pragma unroll
        for (int r2 = 15; r2 > r; --r2) sacc -= sW[r * 52 + r2] * sol[r2];
        sol[r] = sacc / sW[r * 52 + r];
      }
#pragma unroll
      for (int r = 0; r < 16; ++r) sSol[r * 36 + cc2] = sol[r];
    }
    __syncthreads();

    // ---- K = -sol[:, :32], k = -sol[:, 32]; spill gains to workspace ----
    {
      float* KtW = wsK + ((size_t)b * HHd + tcur) * (NUd * NXd);
      for (int idx = lane; idx < NUd * NXd; idx += 32) {
        int r = idx >> 5, j = idx & 31;
        float kv = -sSol[r * 36 + j];
        sK[r * NXd + j] = kv;
        KtW[idx] = kv;
      }
      if (lane < NUd) {
        float kv = -sSol[lane * 36 + 32];
        skv[lane] = kv;
        wsk[((size_t)b * HHd + tcur) * NUd + lane] = kv;
      }
    }
    __syncthreads();

    // ---- Vn = Qxx + Qxu * K : (32x32), 2x2 tiles, K=16 ----
    for (int mt = 0; mt < 2; ++mt)
      for (int nt = 0; nt < 2; ++nt) {
        v8f acc;
#pragma unroll
        for (int i = 0; i < 8; ++i)
          acc[i] = sQ[(mt * 16 + i + hi8) * NNd + nt * 16 + mlo];  // Qxx
#pragma unroll
        for (int kc = 0; kc < 4; ++kc) {
          v2f a, bb;
          a.x  = sQ[(mt * 16 + mlo) * NNd + NXd + kc * 4 + khi];   // Qxu
          a.y  = sQ[(mt * 16 + mlo) * NNd + NXd + kc * 4 + khi + 1];
          bb.x = sK[(kc * 4 + khi) * NXd + nt * 16 + mlo];
          bb.y = sK[(kc * 4 + khi + 1) * NXd + nt * 16 + mlo];
          acc = wmma_f32(a, bb, acc);
        }
#pragma unroll
        for (int i = 0; i < 8; ++i)
          sVn[(mt * 16 + i + hi8) * NXd + nt * 16 + mlo] = acc[i];
      }

    // ---- vn = qx + Qxu @ k (compute into reg before barrier) ----
    float vnval = sq[lane];
#pragma unroll
    for (int u = 0; u < NUd; ++u) vnval += sQ[lane * NNd + NXd + u] * skv[u];
    __syncthreads();  // sVn writes complete

    sv[lane] = vnval;
    // ---- symmetrize: V = 0.5*(Vn + Vn^T) ----
    for (int idx = lane; idx < NXd * NXd; idx += 32) {
      int i = idx >> 5, j = idx & 31;
      sV[idx] = 0.5f * (sVn[i * NXd + j] + sVn[j * NXd + i]);
    }
    __syncthreads();
  }

  // ================= forward closed-loop rollout =================
  sxu[lane] = x0g[(size_t)b * NXd + lane];
  __syncthreads();
  float* Xout = outg + (size_t)b * (HHd + 1) * NXd;
  float* Uout = outg + (size_t)BBd * (HHd + 1) * NXd + (size_t)b * HHd * NUd;
  for (int t = 0; t < HHd; ++t) {
    Xout[t * NXd + lane] = sxu[lane];
    const float* Kt = wsK + ((size_t)b * HHd + t) * (NUd * NXd);
    const float* kt = wsk + ((size_t)b * HHd + t) * NUd;
    if (lane < NUd) {
      float u = kt[lane];
#pragma unroll
      for (int x = 0; x < NXd; ++x) u += Kt[lane * NXd + x] * sxu[x];
      sxu[NXd + lane] = u;
      Uout[t * NUd + lane] = u;
    }
    __syncthreads();
    // x_{t+1} = A x + Bm u = F * [x; u]
    float xn = 0.0f;
#pragma unroll
    for (int cidx = 0; cidx < NNd; ++cidx) xn += sF[lane * NNd + cidx] * sxu[cidx];
    __syncthreads();
    sxu[lane] = xn;
    __syncthreads();
  }
  Xout[HHd * NXd + lane] = sxu[lane];
}

extern "C" void kernel_launch(void* const* d_in, const int* in_sizes, int n_in,
                              void* d_out, int out_size, void* d_ws, size_t ws_size,
                              hipStream_t stream) {
  (void)in_sizes; (void)n_in; (void)out_size; (void)ws_size;
  const float* x0 = (const float*)d_in[0];
  const float* C  = (const float*)d_in[1];
  const float* c  = (const float*)d_in[2];
  const float* Cf = (const float*)d_in[3];
  const float* cf = (const float*)d_in[4];
  const float* A  = (const float*)d_in[5];
  const float* Bm = (const float*)d_in[6];
  float* out = (float*)d_out;
  // workspace: gains K [B*H*NU*NX] then k [B*H*NU]  (~108 MB)
  float* wsK = (float*)d_ws;
  float* wsk = wsK + (size_t)BBd * HHd * NUd * NXd;
  lqr_kernel<<<BBd, 32, 0, stream>>>(x0, C, c, Cf, cf, A, Bm, out, wsK, wsk);
}